// EarthMoversDistanceLoss_36172214567796
// MI455X (gfx1250) — compile-verified
//
#include <hip/hip_runtime.h>

// EMD loss: out[b] = sum_t (cumsum(x)[b,t] - cumsum(y)[b,t])^2
// Memory-bound: 256MB @ 23.3TB/s ~= 11us floor. fp32 throughout.
// Within-chunk cumsum  = A_chunk x triangular-ones  (V_WMMA_F32_16X16X4_F32, reset per chunk)
// Running row prefixes = A_chunk x all-ones         (same WMMA op, accumulated, C-layout,
//                        replicated across columns -> zero cross-lane traffic in the loop)

typedef __attribute__((ext_vector_type(2))) float v2f;
typedef __attribute__((ext_vector_type(8))) float v8f;

#define BINS 256
#define ROWS_PER_WAVE 16
#define WAVES_PER_BLOCK 8
#define ROWS_PER_BLOCK (ROWS_PER_WAVE * WAVES_PER_BLOCK)

__global__ __launch_bounds__(256)
void emd_wmma_kernel(const float* __restrict__ x,
                     const float* __restrict__ y,
                     float* __restrict__ out) {
    const int lane = threadIdx.x & 31;
    const int n    = lane & 15;   // A-matrix row M == C-matrix column N for this lane
    const int h    = lane >> 4;   // lane half: 0 or 1

    const int rowBaseWave = blockIdx.x * ROWS_PER_BLOCK + (threadIdx.x >> 5) * ROWS_PER_WAVE;
    const long row = (long)rowBaseWave + n;          // this lane's A-matrix row
    const float* __restrict__ px = x + row * BINS;
    const float* __restrict__ py = y + row * BINS;

    // Triangular-ones B slices (j-invariant): B[k,n] = (n >= 4q + k), k = 2h + p.
    v2f bq[4];
    #pragma unroll
    for (int q = 0; q < 4; ++q) {
        bq[q].x = (n >= q * 4 + 2 * h)     ? 1.0f : 0.0f;
        bq[q].y = (n >= q * 4 + 2 * h + 1) ? 1.0f : 0.0f;
    }
    v2f bones;                                       // all-ones B (prefix carrier)
    bones.x = 1.0f;
    bones.y = 1.0f;

    // acc[v]: this column's private running sum of cdf^2 for row m = 8h+v.
    // c2[v] : running prefix sum_{previous chunks} d[m, :], replicated over columns.
    float acc[8];
    #pragma unroll
    for (int v = 0; v < 8; ++v) acc[v] = 0.0f;
    v8f c2 = {};

    #pragma unroll 2
    for (int j = 0; j < 16; ++j) {                   // 16 chunks of 16 bins
        // Branchless in-row prefetch two chunks ahead (wraps to row start at the
        // tail -- harmless L0 hit, no per-iteration branch).
        const int pfcol = ((j * 16 + 32) & (BINS - 1)) + 2 * h;
        __builtin_prefetch(px + pfcol, 0, 1);        // global_prefetch_b8
        __builtin_prefetch(py + pfcol, 0, 1);

        // Issue all 4 k-slice loads up front, then the WMMAs.
        v2f d[4];
        #pragma unroll
        for (int q = 0; q < 4; ++q) {
            const int col = j * 16 + q * 4 + 2 * h;  // lane's K = 2h+p within slice
            v2f vx = *(const v2f*)(px + col);        // global_load_b64
            v2f vy = *(const v2f*)(py + col);        // global_load_b64
            d[q].x = vx.x - vy.x;
            d[q].y = vx.y - vy.y;
        }

        v8f c = {};                                  // within-chunk cumsum (fresh)
        #pragma unroll
        for (int q = 0; q < 4; ++q) {
            c = __builtin_amdgcn_wmma_f32_16x16x4_f32(
                    false, d[q], false, bq[q], (short)0, c, false, false);
        }

        // cdf at (row 8h+v, column n) = c[v] + c2[v]; accumulate square privately.
        #pragma unroll
        for (int v = 0; v < 8; ++v) {
            float t = c[v] + c2[v];
            acc[v] = fmaf(t, t, acc[v]);
        }

        // Advance running prefixes: c2 += row-sums of this chunk (B = ones).
        #pragma unroll
        for (int q = 0; q < 4; ++q) {
            c2 = __builtin_amdgcn_wmma_f32_16x16x4_f32(
                    false, d[q], false, bones, (short)0, c2, false, false);
        }
    }

    // One final 16-lane butterfly per VGPR: sum over columns n of each half.
    #pragma unroll
    for (int v = 0; v < 8; ++v) {
        float s = acc[v];
        s += __shfl_xor(s, 1);
        s += __shfl_xor(s, 2);
        s += __shfl_xor(s, 4);
        s += __shfl_xor(s, 8);
        acc[v] = s;
    }

    if (n == 0) {                                    // lanes 0 and 16 write 8 rows each
        #pragma unroll
        for (int v = 0; v < 8; ++v) {
            out[rowBaseWave + 8 * h + v] = acc[v];
        }
    }
}

extern "C" void kernel_launch(void* const* d_in, const int* in_sizes, int n_in,
                              void* d_out, int out_size, void* d_ws, size_t ws_size,
                              hipStream_t stream) {
    (void)n_in; (void)d_ws; (void)ws_size; (void)out_size;
    const float* x = (const float*)d_in[0];
    const float* y = (const float*)d_in[1];
    float* out = (float*)d_out;

    const int rows = in_sizes[0] / BINS;             // 131072 for the reference shapes
    const int blocks = rows / ROWS_PER_BLOCK;        // rows is a multiple of 128 here
    emd_wmma_kernel<<<dim3(blocks), dim3(256), 0, stream>>>(x, y, out);
}